// BasicBlock_21131239097119
// MI455X (gfx1250) — compile-verified
//
#include <hip/hip_runtime.h>
#include <math.h>

typedef __attribute__((ext_vector_type(2))) float v2f;
typedef __attribute__((ext_vector_type(8))) float v8f;

#define C_CH 128
#define HW   28
#define PIX  784            // 28*28
#define NB   32
#define NPIX (NB*PIX)       // 25088
#define TOT  (NB*C_CH*PIX)  // 3,211,264 elements

// ======================= Conv 3x3 (pad 1) via f32 WMMA =======================
// Implicit GEMM: O[co, pix] = sum_{kk,ci} W[co,ci,kk] * X[ci, shift_kk(pix)]
// Block: 256 thr = 8 waves. Block tile: M = 128 (all co) x NP = 64 pixels.
// Wave w owns co rows [16w,16w+16) and 4 accumulators (4 pixel sub-tiles of 16).
#define CI_T 32
#define NP   64
#define LDK  (CI_T + 2)     // padded k-stride in floats -> conflict-free b64 LDS reads

__global__ __launch_bounds__(256)
void conv3x3_wmma(const float* __restrict__ x, const float* __restrict__ wgt,
                  float* __restrict__ out)
{
  __shared__ float Ws[C_CH * LDK];   // weights  [m][k]   128 x 34
  __shared__ float Xs[NP  * LDK];    // activ.   [j][k]    64 x 34

  const int t    = threadIdx.x;
  const int wave = t >> 5;
  const int lane = t & 31;
  const int hl   = lane >> 4;        // half of wave (selects K pair / M+8 rows)
  const int l16  = lane & 15;
  const int m_base = wave * 16;
  const int p0   = blockIdx.x * NP;

  // loader role: thread t fills pixel j = t&63 with k-range [8*(t>>6), +8)
  const int jl   = t & (NP - 1);
  const int kset = t >> 6;           // 0..3
  const int pj = p0 + jl;
  const int nj = pj / PIX;
  const int hj = (pj % PIX) / HW;
  const int wj = pj % HW;

  v8f acc0 = {}, acc1 = {}, acc2 = {}, acc3 = {};

  for (int kk = 0; kk < 9; ++kk) {
    const int dh = kk / 3 - 1, dw = kk % 3 - 1;
    const int hh = hj + dh, ww = wj + dw;
    const bool ok = ((unsigned)hh < HW) & ((unsigned)ww < HW);
    const float* xsrc = x + ((size_t)nj * C_CH) * PIX + hh * HW + ww;

    for (int ci0 = 0; ci0 < C_CH; ci0 += CI_T) {
      __syncthreads();
      { // weights: thread -> m = t&127, k half = 16*(t>>7); stride-9 gathers (L2 resident)
        const int m = t & 127, kh = (t >> 7) * 16;
        const float* wp = wgt + ((size_t)m * C_CH + ci0 + kh) * 9 + kk;
        #pragma unroll
        for (int k = 0; k < 16; ++k)
          Ws[m * LDK + kh + k] = wp[(size_t)k * 9];
      }
      { // activations, zero-padded halo
        #pragma unroll
        for (int k8 = 0; k8 < 8; ++k8) {
          const int k = kset * 8 + k8;
          float v = 0.f;
          if (ok) v = xsrc[(size_t)(ci0 + k) * PIX];
          Xs[jl * LDK + k] = v;
        }
      }
      __syncthreads();

      #pragma unroll
      for (int ks = 0; ks < CI_T / 4; ++ks) {
        const int k0 = ks * 4 + hl * 2;     // A/B f32 frag: lanes<16 K{0,1}, lanes>=16 K{2,3}
        v2f a  = *(const v2f*)&Ws[(m_base + l16) * LDK + k0];
        v2f b0 = *(const v2f*)&Xs[( 0 + l16) * LDK + k0];
        v2f b1 = *(const v2f*)&Xs[(16 + l16) * LDK + k0];
        v2f b2 = *(const v2f*)&Xs[(32 + l16) * LDK + k0];
        v2f b3 = *(const v2f*)&Xs[(48 + l16) * LDK + k0];
        acc0 = __builtin_amdgcn_wmma_f32_16x16x4_f32(false, a, false, b0, (short)0, acc0, false, false);
        acc1 = __builtin_amdgcn_wmma_f32_16x16x4_f32(false, a, false, b1, (short)0, acc1, false, false);
        acc2 = __builtin_amdgcn_wmma_f32_16x16x4_f32(false, a, false, b2, (short)0, acc2, false, false);
        acc3 = __builtin_amdgcn_wmma_f32_16x16x4_f32(false, a, false, b3, (short)0, acc3, false, false);
      }
    }
  }

  // C/D layout: VGPR r, lanes 0-15 -> M = m_base+r, lanes 16-31 -> M = m_base+8+r; N = lane%16
  const int mrow = m_base + hl * 8;
  #pragma unroll
  for (int r = 0; r < 8; ++r) {
    #pragma unroll
    for (int jt = 0; jt < 4; ++jt) {
      const int p = p0 + jt * 16 + l16;
      const int n = p / PIX, hp = p % PIX;
      float v = (jt == 0) ? acc0[r] : (jt == 1) ? acc1[r] : (jt == 2) ? acc2[r] : acc3[r];
      out[((size_t)(n * C_CH + mrow + r)) * PIX + hp] = v;
    }
  }
}

// ======================= AdderNet 3x3 (pure VALU, LDS tiled) =======================
// out[n,co,h,w] = -sum_{ci,kh,kw} |in[n,ci,h+kh-1,w+kw-1] - w[co,ci,kh,kw]|
// Block 224 thr (7 waves): thread = (co_l 0..7, w 0..27); owns co_l and co_l+8,
// 7 output rows each -> 16co x 28w x 7h register tile per block. ci chunks of 16.
#define ACI 16
#define TPB_ADD 224

__global__ __launch_bounds__(TPB_ADD)
void adder3x3(const float* __restrict__ in, const float* __restrict__ wgt,
              float* __restrict__ out)
{
  __shared__ float Xt[ACI * 9 * 32];    // [k][row 0..8][col 0..29, pad 32]
  __shared__ float Wt[16 * ACI * 9];    // [cc][k][q]

  const int t    = threadIdx.x;
  const int w    = t % HW;              // 0..27
  const int co_l = t / HW;              // 0..7
  const int hb      = blockIdx.x * 7;   // 4 h-blocks of 7 rows
  const int co_base = blockIdx.y * 16;  // 8 co-tiles
  const int n       = blockIdx.z;       // batch

  float acc0[7], acc1[7];
  #pragma unroll
  for (int r = 0; r < 7; ++r) { acc0[r] = 0.f; acc1[r] = 0.f; }

  for (int ci0 = 0; ci0 < C_CH; ci0 += ACI) {
    __syncthreads();
    // input tile: rows hb-1..hb+7, cols -1..28, zero-padded halo
    for (int idx = t; idx < ACI * 9 * 30; idx += TPB_ADD) {
      const int k = idx / 270, rr = idx % 270, r = rr / 30, col = rr % 30;
      const int hh = hb - 1 + r, ww = col - 1;
      float v = 0.f;
      if (((unsigned)hh < HW) & ((unsigned)ww < HW))
        v = in[((size_t)(n * C_CH + ci0 + k)) * PIX + hh * HW + ww];
      Xt[(k * 9 + r) * 32 + col] = v;
    }
    // weight tile (contiguous copy)
    for (int idx = t; idx < 16 * ACI * 9; idx += TPB_ADD) {
      const int cc = idx / (ACI * 9), rem = idx % (ACI * 9);
      Wt[idx] = wgt[((size_t)((co_base + cc) * C_CH + ci0)) * 9 + rem];
    }
    __syncthreads();

    for (int k = 0; k < ACI; ++k) {
      float wA[9], wB[9];
      #pragma unroll
      for (int q = 0; q < 9; ++q) {        // LDS broadcasts across the 28 w-threads
        wA[q] = Wt[((co_l    ) * ACI + k) * 9 + q];
        wB[q] = Wt[((co_l + 8) * ACI + k) * 9 + q];
      }
      const float* xr = &Xt[(k * 9) * 32 + w];   // patch cols = w..w+2 (halo shift)
      float win[3][3];
      #pragma unroll
      for (int r = 0; r < 2; ++r)
        #pragma unroll
        for (int c2 = 0; c2 < 3; ++c2) win[r][c2] = xr[r * 32 + c2];
      #pragma unroll
      for (int ro = 0; ro < 7; ++ro) {     // sliding 3-row window down 7 outputs
        const int r2 = (ro + 2) % 3;
        #pragma unroll
        for (int c2 = 0; c2 < 3; ++c2) win[r2][c2] = xr[(ro + 2) * 32 + c2];
        float sA = 0.f, sB = 0.f;
        #pragma unroll
        for (int kr = 0; kr < 3; ++kr) {
          const int rb = (ro + kr) % 3;
          #pragma unroll
          for (int c2 = 0; c2 < 3; ++c2) {
            sA += fabsf(win[rb][c2] - wA[kr * 3 + c2]);   // v_sub + v_sub|abs|
            sB += fabsf(win[rb][c2] - wB[kr * 3 + c2]);
          }
        }
        acc0[ro] += sA; acc1[ro] += sB;
      }
    }
  }

  #pragma unroll
  for (int ro = 0; ro < 7; ++ro) {
    out[((size_t)(n * C_CH + co_base + co_l    )) * PIX + (hb + ro) * HW + w] = -acc0[ro];
    out[((size_t)(n * C_CH + co_base + co_l + 8)) * PIX + (hb + ro) * HW + w] = -acc1[ro];
  }
}

// ======================= BatchNorm (training) =======================
__global__ __launch_bounds__(256)
void bn_stats(const float* __restrict__ in, float* __restrict__ stats)
{
  __shared__ float s1[256], s2[256];
  const int c = blockIdx.x, t = threadIdx.x;
  float s = 0.f, q = 0.f;
  for (int i = t; i < NB * PIX; i += 256) {
    const int nn = i / PIX, p = i % PIX;
    const float v = in[((size_t)(nn * C_CH + c)) * PIX + p];
    s += v; q += v * v;
  }
  s1[t] = s; s2[t] = q;
  __syncthreads();
  for (int off = 128; off > 0; off >>= 1) {
    if (t < off) { s1[t] += s1[t + off]; s2[t] += s2[t + off]; }
    __syncthreads();
  }
  if (t == 0) {
    const float inv_m = 1.f / (float)(NB * PIX);
    const float mean = s1[0] * inv_m;
    const float var  = s2[0] * inv_m - mean * mean;   // biased var (matches jnp.var)
    stats[c]        = mean;
    stats[C_CH + c] = rsqrtf(var + 1e-5f);
  }
}

__global__ __launch_bounds__(256)
void bn_apply(const float* __restrict__ in, const float* __restrict__ stats,
              const float* __restrict__ gamma, const float* __restrict__ beta,
              const float* __restrict__ residual, float* __restrict__ out)
{
  const int idx = blockIdx.x * 256 + threadIdx.x;
  if (idx >= TOT) return;
  const int c = (idx / PIX) % C_CH;
  float v = (in[idx] - stats[c]) * stats[C_CH + c] * gamma[c] + beta[c];
  if (residual) v += residual[idx];
  out[idx] = fmaxf(v, 0.f);
}

// ======================= Launch =======================
extern "C" void kernel_launch(void* const* d_in, const int* in_sizes, int n_in,
                              void* d_out, int out_size, void* d_ws, size_t ws_size,
                              hipStream_t stream) {
  const float* x       = (const float*)d_in[0];
  const float* conv1_w = (const float*)d_in[1];
  const float* add1_w  = (const float*)d_in[2];
  const float* bn1_g   = (const float*)d_in[3];
  const float* bn1_b   = (const float*)d_in[4];
  const float* conv2_w = (const float*)d_in[5];
  const float* add2_w  = (const float*)d_in[6];
  const float* bn2_g   = (const float*)d_in[7];
  const float* bn2_b   = (const float*)d_in[8];
  float* out = (float*)d_out;

  float* A   = (float*)d_ws;      // ping
  float* B   = A + TOT;           // pong
  float* st1 = B + TOT;           // 256 floats: mean | inv_std
  float* st2 = st1 + 256;

  const dim3 addGrid(4, 8, NB);
  const int bnBlocks = (TOT + 255) / 256;

  conv3x3_wmma<<<NPIX / NP, 256, 0, stream>>>(x, conv1_w, A);
  adder3x3    <<<addGrid, TPB_ADD, 0, stream>>>(A, add1_w, B);
  bn_stats    <<<C_CH, 256, 0, stream>>>(B, st1);
  bn_apply    <<<bnBlocks, 256, 0, stream>>>(B, st1, bn1_g, bn1_b, nullptr, A);

  conv3x3_wmma<<<NPIX / NP, 256, 0, stream>>>(A, conv2_w, B);
  adder3x3    <<<addGrid, TPB_ADD, 0, stream>>>(B, add2_w, A);
  bn_stats    <<<C_CH, 256, 0, stream>>>(A, st2);
  bn_apply    <<<bnBlocks, 256, 0, stream>>>(A, st2, bn2_g, bn2_b, x, out);
}